// HashEncoderWithPositional_88364657148057
// MI455X (gfx1250) — compile-verified
//
#include <hip/hip_runtime.h>
#include <stdint.h>

// ---------------- static config (mirrors the reference) ----------------
#define N_POINTS        2000000
#define THREADS         128            // 4 wave32s per block
#define OUT_DIM         71             // 32 hash feats + 3 x + 36 sin/cos
#define WAVE_PTS        32
#define WAVE_BYTES      (WAVE_PTS * OUT_DIM * 4)   // 9088, multiple of 16
#define FULL_CHUNK_IT   17             // 17*32 = 544 of 568 16B chunks
#define TAIL_LANES      24             // 568 - 544 = 24 chunks in last iter

// floor(16 * 2^l)
static __device__ constexpr int kRes[16] = {
    16, 32, 64, 128, 256, 512, 1024, 2048, 4096, 8192,
    16384, 32768, 65536, 131072, 262144, 524288};
// cumulative row offsets into the table (levels 0..2 dense, 3..15 hashed 2^19)
static __device__ constexpr uint32_t kOff[16] = {
    0u, 4913u, 40850u, 315475u, 839763u, 1364051u, 1888339u, 2412627u,
    2936915u, 3461203u, 3985491u, 4509779u, 5034067u, 5558355u, 6082643u,
    6606931u};

__global__ __launch_bounds__(THREADS)
void HashEncoderWithPositional_88364657148057_kernel(
    const float* __restrict__ x,      // [N,3]
    const float* __restrict__ table,  // [7131219,2]
    float* __restrict__ out)          // [N,71]
{
    __shared__ float smem[THREADS * OUT_DIM];   // 36,352 B

    const int lane = (int)(threadIdx.x & 31u);
    const int wave = (int)(threadIdx.x >> 5u);
    const int n    = (int)(blockIdx.x * THREADS + threadIdx.x);

    const float xv = x[3 * n + 0];
    const float yv = x[3 * n + 1];
    const float zv = x[3 * n + 2];

    float* row = &smem[threadIdx.x * OUT_DIM];
    const float2* __restrict__ t2 = (const float2*)table;

    // ---------------- hash-grid features (16 levels, unrolled) ----------------
#pragma unroll
    for (int l = 0; l < 16; ++l) {
        const int   res = kRes[l];
        const float fr  = (float)res;

        const float px = xv * fr, py = yv * fr, pz = zv * fr;
        const float fx = floorf(px), fy = floorf(py), fz = floorf(pz);
        const float tx = px - fx,   ty = py - fy,   tz = pz - fz;
        const int ix = (int)fx, iy = (int)fy, iz = (int)fz;

        // clip(p0 + {0,1}, 0, res) — matches reference
        const int x0 = max(min(ix,     res), 0), x1 = max(min(ix + 1, res), 0);
        const int y0 = max(min(iy,     res), 0), y1 = max(min(iy + 1, res), 0);
        const int z0 = max(min(iz,     res), 0), z1 = max(min(iz + 1, res), 0);

        const float wx0 = 1.0f - tx, wx1 = tx;
        const float wy0 = 1.0f - ty, wy1 = ty;
        const float wz0 = 1.0f - tz, wz1 = tz;

        float f0 = 0.0f, f1 = 0.0f;
#pragma unroll
        for (int c = 0; c < 8; ++c) {           // corner bits: [x,y,z] = [c&4,c&2,c&1]
            const int cx = (c & 4) ? x1 : x0;
            const int cy = (c & 2) ? y1 : y0;
            const int cz = (c & 1) ? z1 : z0;
            uint32_t idx;
            if (l < 3) {                        // dense level: stride indexing
                const int r1 = res + 1;
                idx = (uint32_t)(cx + cy * r1 + cz * r1 * r1);
            } else {                            // spatial hash, size = 2^19
                const uint32_t h = (uint32_t)cx * 1u
                                 ^ (uint32_t)cy * 2654435761u
                                 ^ (uint32_t)cz * 805459861u;
                idx = h & 0x7FFFFu;
            }
            const float2 v = t2[kOff[l] + idx]; // L2-resident gather
            const float  w = ((c & 4) ? wx1 : wx0)
                           * ((c & 2) ? wy1 : wy0)
                           * ((c & 1) ? wz1 : wz0);
            f0 = fmaf(w, v.x, f0);
            f1 = fmaf(w, v.y, f1);
        }
        row[2 * l + 0] = f0;
        row[2 * l + 1] = f1;
    }

    // ---------------- positional encoding ----------------
    row[32] = xv; row[33] = yv; row[34] = zv;

    float sx, cxv, sy, cyv, sz, czv;
    sincosf(xv, &sx, &cxv);
    sincosf(yv, &sy, &cyv);
    sincosf(zv, &sz, &czv);

#pragma unroll
    for (int f = 0; f < 6; ++f) {
        float* p = row + 35 + f * 6;            // [sin xyz, cos xyz] per octave
        p[0] = sx;  p[1] = sy;  p[2] = sz;
        p[3] = cxv; p[4] = cyv; p[5] = czv;
        // double-angle: sin(2a)=2 sin a cos a ; cos(2a)=1-2 sin^2 a
        const float nsx = 2.0f * sx * cxv, ncx = fmaf(-2.0f * sx, sx, 1.0f);
        const float nsy = 2.0f * sy * cyv, ncy = fmaf(-2.0f * sy, sy, 1.0f);
        const float nsz = 2.0f * sz * czv, ncz = fmaf(-2.0f * sz, sz, 1.0f);
        sx = nsx; cxv = ncx; sy = nsy; cyv = ncy; sz = nsz; czv = ncz;
    }

    // Wait for this wave's own LDS stores to land before the async engine
    // reads them (per-wave tile, no cross-wave sharing -> no barrier needed).
    asm volatile("s_wait_dscnt 0x0" ::: "memory");

    // ---------------- coalesced LDS -> global flush (CDNA5 async path) ------
    // Wave tile: 32 rows * 284 B = 9088 B contiguous in both LDS and memory,
    // 16B aligned (9088 % 16 == 0). 512 B per instruction across the wave.
    const uint64_t gbase =
        (uint64_t)(uintptr_t)out
        + ((uint64_t)blockIdx.x * THREADS + (uint64_t)wave * WAVE_PTS)
              * (uint64_t)(OUT_DIM * 4)
        + (uint64_t)((uint32_t)lane * 16u);
    // low 32 bits of a flat LDS pointer == workgroup-relative LDS byte offset
    const uint32_t lbase =
        (uint32_t)(uintptr_t)&smem[wave * WAVE_PTS * OUT_DIM]
        + (uint32_t)lane * 16u;

#pragma unroll
    for (int k = 0; k < FULL_CHUNK_IT + 1; ++k) {
        if (k < FULL_CHUNK_IT || lane < TAIL_LANES) {
            const uint64_t ga = gbase + (uint32_t)(k * 512);
            const uint32_t la = lbase + (uint32_t)(k * 512);
            // Non-temporal: don't let 568 MB of streaming output evict the
            // 57 MB table out of the 192 MB L2.
            asm volatile(
                "global_store_async_from_lds_b128 %0, %1, off th:TH_STORE_NT"
                :: "v"(ga), "v"(la) : "memory");
        }
    }
    __builtin_amdgcn_s_wait_asynccnt(0);
}

extern "C" void kernel_launch(void* const* d_in, const int* in_sizes, int n_in,
                              void* d_out, int out_size, void* d_ws, size_t ws_size,
                              hipStream_t stream) {
    (void)in_sizes; (void)n_in; (void)out_size; (void)d_ws; (void)ws_size;
    const float* x     = (const float*)d_in[0];   // [2,000,000 * 3]
    const float* table = (const float*)d_in[1];   // [7,131,219 * 2]
    float*       out   = (float*)d_out;           // [2,000,000 * 71]

    const int blocks = N_POINTS / THREADS;        // 15625, exact
    HashEncoderWithPositional_88364657148057_kernel
        <<<blocks, THREADS, 0, stream>>>(x, table, out);
}